// CapChMatch_5841155523251
// MI455X (gfx1250) — compile-verified
//
#include <hip/hip_runtime.h>

typedef float v2f __attribute__((ext_vector_type(2)));
typedef float v8f __attribute__((ext_vector_type(8)));

#define B0N 2
#define B1N 4
#define CN  64
#define HN  32
#define WN  32
#define KLEN 9
#define SN  8
#define PN  1024      // H*W
#define WPAD 34       // W + 2*pad
#define N1  36        // B1N*KLEN
#define JBLOCKS 64
#define JCHUNK  2048  // 131072 / 64

// ---------------------------------------------------------------------------
// Pass 1a: T[b1,k,c] = sum over (b0,p) of gathered x.  grid = N1*CN blocks.
// ---------------------------------------------------------------------------
__global__ void k_reduce_g(const float* __restrict__ x,
                           const int* __restrict__ indexm,
                           float* __restrict__ T) {
  __shared__ float red[256];
  const int bid = blockIdx.x;          // n1*64 + c
  const int c   = bid & 63;
  const int n1  = bid >> 6;
  const int k   = n1 % KLEN;
  const int b1  = n1 / KLEN;
  const int t   = threadIdx.x;
  float sum = 0.f;
  #pragma unroll
  for (int i = 0; i < 8; ++i) {
    const int e  = t + i * 256;        // e in [0, 2048): (b0,p)
    const int b0 = e >> 10;
    const int p  = e & 1023;
    const int idx = indexm[k * PN + p];
    const int rr = idx / WPAD;
    const int cc = idx - rr * WPAD;
    const int hr = rr - 1, wc = cc - 1;
    if ((unsigned)hr < (unsigned)HN && (unsigned)wc < (unsigned)WN)
      sum += x[(((b0 * B1N) + b1) * CN + c) * PN + hr * WN + wc];
  }
  red[t] = sum;
  __syncthreads();
  for (int st = 128; st > 0; st >>= 1) {
    if (t < st) red[t] += red[t + st];
    __syncthreads();
  }
  if (t == 0) T[bid] = red[0];
}

// ---------------------------------------------------------------------------
// Pass 1b: s1[(b1,k),s] = (1/8) * sum_c T[b1,k,c] * w[c,k,s]
// One wave per k. Batched 4x64x8 GEMM via V_WMMA_F32_16X16X4_F32 (16 K-steps).
// A 16x4 layout: lanes 0-15 hold M=row, vgpr0/1 = K0/K1; lanes 16-31 = K2/K3.
// B 4x16 layout mirrors (lanes = N column, vgpr0/1 = rows K0/K1 | K2/K3).
// D: vgpr r -> lanes 0-15 M=r, N=lane.
// ---------------------------------------------------------------------------
__global__ void k_gemm_s1(const float* __restrict__ T,
                          const float* __restrict__ w,
                          float* __restrict__ s1) {
  const int k    = blockIdx.x;
  const int lane = threadIdx.x;
  const int mn   = lane & 15;
  const bool hi  = lane >= 16;
  v8f acc = {0.f,0.f,0.f,0.f,0.f,0.f,0.f,0.f};
  for (int ci = 0; ci < 16; ++ci) {
    const int c0 = ci * 4 + (hi ? 2 : 0);
    float a0 = 0.f, a1 = 0.f, b0 = 0.f, b1 = 0.f;
    if (mn < B1N) {                       // A row = b1
      a0 = T[(mn * KLEN + k) * CN + c0];
      a1 = T[(mn * KLEN + k) * CN + c0 + 1];
    }
    if (mn < SN) {                        // B col = s
      b0 = w[(c0 * KLEN + k) * SN + mn];
      b1 = w[((c0 + 1) * KLEN + k) * SN + mn];
    }
    v2f A = {a0, a1};
    v2f B = {b0, b1};
    acc = __builtin_amdgcn_wmma_f32_16x16x4_f32(
        false, A, false, B, (short)0, acc, false, false);
  }
  if (!hi && mn < SN) {
    #pragma unroll
    for (int r = 0; r < B1N; ++r)         // D row r = b1
      s1[(r * KLEN + k) * SN + mn] = acc[r] * (1.0f / SN);
  }
}

// ---------------------------------------------------------------------------
// Routing pass: partial[n1, jb, s] = sum_{j in block} u * softmax_s(u * v)
// grid = (JBLOCKS, N1), 256 threads, 8 j's per thread.
// ---------------------------------------------------------------------------
__global__ void k_route(const float* __restrict__ x,
                        const float* __restrict__ w,
                        const int* __restrict__ indexm,
                        const float* __restrict__ v,
                        float* __restrict__ partial) {
  __shared__ float wsm[CN * SN];          // w[:, k, :] staged once (2 KB)
  __shared__ float vv[SN];
  __shared__ float red[256 * 9];          // padded to avoid bank conflicts
  const int n1 = blockIdx.y;
  const int k  = n1 % KLEN;
  const int b1 = n1 / KLEN;
  const int t  = threadIdx.x;

  for (int i = t; i < CN * SN; i += 256) {
    const int c = i >> 3, s = i & 7;
    wsm[i] = w[(c * KLEN + k) * SN + s];
  }
  if (t < SN) vv[t] = v[n1 * SN + t];
  __syncthreads();

  float vr[SN];
  #pragma unroll
  for (int s = 0; s < SN; ++s) vr[s] = vv[s];
  float acc[SN];
  #pragma unroll
  for (int s = 0; s < SN; ++s) acc[s] = 0.f;

  const int jbase = blockIdx.x * JCHUNK;
  for (int i = 0; i < 8; ++i) {
    const int j  = jbase + t + i * 256;   // j = (b0, c, p)
    const int b0 = j >> 16;
    const int c  = (j >> 10) & 63;
    const int p  = j & 1023;
    const int idx = indexm[k * PN + p];
    const int rr = idx / WPAD;
    const int cc = idx - rr * WPAD;
    const int hr = rr - 1, wc = cc - 1;
    float xv = 0.f;
    if ((unsigned)hr < (unsigned)HN && (unsigned)wc < (unsigned)WN)
      xv = x[(((b0 * B1N) + b1) * CN + c) * PN + hr * WN + wc];

    float us[SN], es[SN];
    float m = -3.4e38f;
    #pragma unroll
    for (int s = 0; s < SN; ++s) {
      us[s] = xv * wsm[c * SN + s];
      es[s] = us[s] * vr[s];
      m = fmaxf(m, es[s]);
    }
    float se = 0.f;
    #pragma unroll
    for (int s = 0; s < SN; ++s) { es[s] = __expf(es[s] - m); se += es[s]; }
    const float inv = 1.0f / se;
    #pragma unroll
    for (int s = 0; s < SN; ++s) acc[s] += us[s] * (es[s] * inv);
  }

  #pragma unroll
  for (int s = 0; s < SN; ++s) red[t * 9 + s] = acc[s];
  __syncthreads();
  for (int st = 128; st > 0; st >>= 1) {
    if (t < st) {
      #pragma unroll
      for (int s = 0; s < SN; ++s) red[t * 9 + s] += red[(t + st) * 9 + s];
    }
    __syncthreads();
  }
  if (t < SN)
    partial[(n1 * gridDim.x + blockIdx.x) * SN + t] = red[t];  // row0: red[0*9+t]
}

// ---------------------------------------------------------------------------
// Reduce partials over jblocks + squash; optionally finalize (v/2 + 0.5).
// grid = N1 blocks of one wave.
// ---------------------------------------------------------------------------
__global__ void k_squash(const float* __restrict__ partial, int nb,
                         float* __restrict__ out, int finalize) {
  __shared__ float sv[SN];
  const int n1 = blockIdx.x;
  const int t  = threadIdx.x;
  if (t < SN) {
    float s = 0.f;
    for (int jb = 0; jb < nb; ++jb) s += partial[(n1 * nb + jb) * SN + t];
    sv[t] = s;
  }
  __syncthreads();
  if (t < SN) {
    float n2 = 0.f;
    #pragma unroll
    for (int i = 0; i < SN; ++i) n2 += sv[i] * sv[i];
    float val = (n2 / (1.0f + n2)) * sv[t] / sqrtf(n2 + 1e-8f);
    out[n1 * SN + t] = finalize ? (val * 0.5f + 0.5f) : val;
  }
}

// ---------------------------------------------------------------------------
extern "C" void kernel_launch(void* const* d_in, const int* in_sizes, int n_in,
                              void* d_out, int out_size, void* d_ws, size_t ws_size,
                              hipStream_t stream) {
  const float* x      = (const float*)d_in[0];   // (2,4,64,32,32)
  const float* w      = (const float*)d_in[1];   // (1,1,64,9,1,8)
  const int*   indexm = (const int*)d_in[2];     // (9216,)
  float* out = (float*)d_out;                    // (36,1,8) -> 288 f32

  float* ws      = (float*)d_ws;
  float* T       = ws;                 // N1*CN          = 2304
  float* s1      = T + N1 * CN;        // N1*SN          = 288
  float* v       = s1 + N1 * SN;       // N1*SN          = 288
  float* partial = v + N1 * SN;        // N1*JBLOCKS*SN  = 18432

  // iter 0 (uniform coupling 1/8): reduction + WMMA contraction + squash
  k_reduce_g<<<dim3(N1 * CN), 256, 0, stream>>>(x, indexm, T);
  k_gemm_s1<<<dim3(KLEN), 32, 0, stream>>>(T, w, s1);
  k_squash<<<dim3(N1), 32, 0, stream>>>(s1, 1, v, 0);
  // iter 1
  k_route<<<dim3(JBLOCKS, N1), 256, 0, stream>>>(x, w, indexm, v, partial);
  k_squash<<<dim3(N1), 32, 0, stream>>>(partial, JBLOCKS, v, 0);
  // final iteration + output transform
  k_route<<<dim3(JBLOCKS, N1), 256, 0, stream>>>(x, w, indexm, v, partial);
  k_squash<<<dim3(N1), 32, 0, stream>>>(partial, JBLOCKS, out, 1);
}